// ProteinGNN_13726715478760
// MI455X (gfx1250) — compile-verified
//
#include <hip/hip_runtime.h>
#include <math.h>

#define HF 64      // hidden features
#define OUTF 128   // output features
#define INF 20     // input features

typedef __attribute__((ext_vector_type(2))) float v2f;
typedef __attribute__((ext_vector_type(8))) float v8f;

// ---------------- degree / normalization ----------------
__global__ void pg_deg_init(unsigned int* deg, int n) {
  int i = blockIdx.x * blockDim.x + threadIdx.x;
  if (i < n) deg[i] = 1u;  // self-loop
}

__global__ void pg_deg_count(const int* __restrict__ dst, unsigned int* deg, int e) {
  int i = blockIdx.x * blockDim.x + threadIdx.x;
  if (i < e) atomicAdd(&deg[dst[i]], 1u);
}

__global__ void pg_dinv(float* dinv, int n) {
  int i = blockIdx.x * blockDim.x + threadIdx.x;
  if (i < n) {
    unsigned int d = __float_as_uint(dinv[i]);  // slot currently holds uint count
    dinv[i] = rsqrtf((float)d);
  }
}

// ---------------- f32 WMMA GEMM: C[n_rows x n_cols] = A[n_rows x K] @ B[K x n_cols]
// One wave computes one 16x16 tile via V_WMMA_F32_16X16X4_F32 (exact fp32).
__global__ void pg_gemm_wmma(const float* __restrict__ A, const float* __restrict__ Bm,
                             float* __restrict__ C, int n_rows, int K, int n_cols,
                             int relu_in) {
  int gid  = blockIdx.x * blockDim.x + threadIdx.x;
  int wave = gid >> 5;
  int lane = threadIdx.x & 31;
  int col_tiles = n_cols >> 4;
  int row_tile  = wave / col_tiles;
  int col_tile  = wave - row_tile * col_tiles;
  int row_base  = row_tile << 4;
  if (row_base >= n_rows) return;          // wave-uniform exit (EXEC stays all-ones for WMMA)
  int col_base = col_tile << 4;
  int l15   = lane & 15;
  int khalf = lane >> 4;                   // 0: K={0,1}, 1: K={2,3} within each k-step
  int row   = row_base + l15;
  bool rok  = row < n_rows;
  const float* Arow = A + (size_t)(rok ? row : 0) * K;

  v8f acc = {};
  for (int k0 = 0; k0 < K; k0 += 4) {
    int ka = k0 + 2 * khalf;
    float a0 = Arow[ka];
    float a1 = Arow[ka + 1];
    if (!rok) { a0 = 0.0f; a1 = 0.0f; }
    if (relu_in) { a0 = fmaxf(a0, 0.0f); a1 = fmaxf(a1, 0.0f); }
    v2f a, b;
    a.x = a0; a.y = a1;
    b.x = Bm[(size_t)ka * n_cols + col_base + l15];
    b.y = Bm[(size_t)(ka + 1) * n_cols + col_base + l15];
    // (neg_a, A, neg_b, B, c_mod, C, reuse_a, reuse_b)
    acc = __builtin_amdgcn_wmma_f32_16x16x4_f32(false, a, false, b, (short)0, acc,
                                                false, false);
  }
#pragma unroll
  for (int v = 0; v < 8; ++v) {
    int r = row_base + v + 8 * khalf;      // C layout: VGPR v -> M = v + 8*khalf
    if (r < n_rows) C[(size_t)r * n_cols + col_base + l15] = acc[v];
  }
}

// ---------------- aggregation: init with self-loop term + bias ----------------
__global__ void pg_agg_init(const float* __restrict__ h, const float* __restrict__ dinv,
                            const float* __restrict__ bias, float* __restrict__ agg, int n) {
  int t = blockIdx.x * blockDim.x + threadIdx.x;
  if (t >= n * HF) return;
  int i = t >> 6;
  float di = dinv[i];
  agg[t] = di * di * h[t] + bias[t & 63];
}

// 64 lanes per edge: coalesced 256B gather from h[src], 256B atomic burst to agg[dst]
__global__ void pg_edge_agg(const int* __restrict__ src, const int* __restrict__ dst,
                            const float* __restrict__ h, const float* __restrict__ dinv,
                            float* __restrict__ agg, int e) {
  int t  = blockIdx.x * blockDim.x + threadIdx.x;
  int ei = t >> 6;
  if (ei >= e) return;
  int j = t & 63;
  int s = src[ei];
  int d = dst[ei];
  float norm = dinv[s] * dinv[d];
  atomicAdd(&agg[(size_t)d * HF + j], norm * h[(size_t)s * HF + j]);
}

// ---------------- mean pool (ReLU fused) ----------------
__global__ void pg_zero64(float* gsum) {
  if (threadIdx.x < HF) gsum[threadIdx.x] = 0.0f;
}

__global__ void pg_mean_pool(const float* __restrict__ agg, float* gsum, int n) {
  int j    = threadIdx.x & 63;
  int grp  = blockIdx.x * (blockDim.x >> 6) + (threadIdx.x >> 6);
  int ngrp = gridDim.x * (blockDim.x >> 6);
  float s = 0.0f;
  for (int i = grp; i < n; i += ngrp) {
    float v = agg[(size_t)i * HF + j];
    s += (v > 0.0f) ? v : 0.0f;
  }
  atomicAdd(&gsum[j], s);
}

// ---------------- FC head: g -> relu(g@fcW1+b) -> sigmoid(.@fcW2+b) ----------------
__global__ void pg_head(const float* __restrict__ gsum,
                        const float* __restrict__ fcW1, const float* __restrict__ fcb1,
                        const float* __restrict__ fcW2, const float* __restrict__ fcb2,
                        float* __restrict__ out, float invN) {
  __shared__ float g[HF];
  __shared__ float g1[HF];
  int t = threadIdx.x;  // 128 threads
  if (t < HF) g[t] = gsum[t] * invN;
  __syncthreads();
  if (t < HF) {
    float acc = fcb1[t];
    for (int k = 0; k < HF; ++k) acc += g[k] * fcW1[k * HF + t];
    g1[t] = (acc > 0.0f) ? acc : 0.0f;
  }
  __syncthreads();
  float acc = fcb2[t];
  for (int k = 0; k < HF; ++k) acc += g1[k] * fcW2[k * OUTF + t];
  out[t] = 1.0f / (1.0f + expf(-acc));
}

// ---------------- launch ----------------
extern "C" void kernel_launch(void* const* d_in, const int* in_sizes, int n_in,
                              void* d_out, int out_size, void* d_ws, size_t ws_size,
                              hipStream_t stream) {
  const float* x    = (const float*)d_in[0];
  const int*   ei   = (const int*)d_in[1];
  const float* W1   = (const float*)d_in[2];
  const float* b1   = (const float*)d_in[3];
  const float* W2   = (const float*)d_in[4];
  const float* b2   = (const float*)d_in[5];
  const float* fcW1 = (const float*)d_in[6];
  const float* fcb1 = (const float*)d_in[7];
  const float* fcW2 = (const float*)d_in[8];
  const float* fcb2 = (const float*)d_in[9];

  int n = in_sizes[0] / INF;
  int e = in_sizes[1] / 2;
  const int* src = ei;
  const int* dst = ei + e;

  float* ws     = (float*)d_ws;
  float* dinv   = ws;                          // n floats (first used as uint degree)
  float* hbuf   = dinv + n;                    // n*64
  float* aggbuf = hbuf + (size_t)n * HF;       // n*64 (layer1 agg, reused for layer2)
  float* gsum   = aggbuf + (size_t)n * HF;     // 64

  int th = 256;
  int nb_n   = (n + th - 1) / th;
  int nb_e   = (e + th - 1) / th;
  int nb_nh  = (n * HF + th - 1) / th;
  int nb_eh  = (int)(((size_t)e * HF + th - 1) / th);

  // symmetric normalization: deg (exact int) -> 1/sqrt(deg)
  pg_deg_init<<<nb_n, th, 0, stream>>>((unsigned int*)dinv, n);
  pg_deg_count<<<nb_e, th, 0, stream>>>(dst, (unsigned int*)dinv, e);
  pg_dinv<<<nb_n, th, 0, stream>>>(dinv, n);

  int row_tiles = (n + 15) / 16;
  int col_tiles = HF / 16;
  int waves  = row_tiles * col_tiles;
  int gblk   = (waves + 7) / 8;                // 8 waves / 256-thread block

  // layer 1: h1 = x @ W1 ; agg1 = D^-1/2 A D^-1/2 h1 + b1
  pg_gemm_wmma<<<gblk, th, 0, stream>>>(x, W1, hbuf, n, INF, HF, 0);
  pg_agg_init<<<nb_nh, th, 0, stream>>>(hbuf, dinv, b1, aggbuf, n);
  pg_edge_agg<<<nb_eh, th, 0, stream>>>(src, dst, hbuf, dinv, aggbuf, e);

  // layer 2: h2 = relu(agg1) @ W2 (ReLU fused on A-load) ; agg2 = norm-agg + b2
  pg_gemm_wmma<<<gblk, th, 0, stream>>>(aggbuf, W2, hbuf, n, HF, HF, 1);
  pg_agg_init<<<nb_nh, th, 0, stream>>>(hbuf, dinv, b2, aggbuf, n);
  pg_edge_agg<<<nb_eh, th, 0, stream>>>(src, dst, hbuf, dinv, aggbuf, e);

  // readout: mean(relu(agg2)) -> FC -> sigmoid
  pg_zero64<<<1, HF, 0, stream>>>(gsum);
  pg_mean_pool<<<1024, th, 0, stream>>>(aggbuf, gsum, n);
  pg_head<<<1, OUTF, 0, stream>>>(gsum, fcW1, fcb1, fcW2, fcb2, (float*)d_out,
                                  1.0f / (float)n);
}